// MultiBoxLoss_42314017800715
// MI455X (gfx1250) — compile-verified
//
#include <hip/hip_runtime.h>
#include <math.h>

// Problem constants (match reference)
#define BB 8
#define PP 131072
#define MM 16
#define PT 8                 // priors per thread
#define BLK 256              // threads per block
#define CHUNK (PT * BLK)     // 2048 priors per block
#define NBLK_X (PP / CHUNK)  // 64 chunks per batch
#define NPARTIAL (NBLK_X * BB)
#define ALPHA_W 0.25f
#define THRESH 0.5f

typedef float v2f __attribute__((ext_vector_type(2)));
typedef float v8f __attribute__((ext_vector_type(8)));

// Exact f32 wave32 all-reduce using V_WMMA_F32_16X16X4_F32.
// A = {val, 0} per lane (16x4 f32), B = all ones (4x16 f32), C = 0.
// D[i][j] = rowsum_i(A)  =>  sum over ALL 256 D elements == 16 * sum(A).
// This identity holds under any lane/VGPR layout permutation, so we sum the
// 8 D registers per lane, butterfly over 32 lanes, and scale by 1/16 (exact).
__device__ __forceinline__ float wave_allreduce_sum(float v) {
  v2f a; a.x = v;    a.y = 0.0f;
  v2f o; o.x = 1.0f; o.y = 1.0f;
  v8f c = {};
  v8f d = __builtin_amdgcn_wmma_f32_16x16x4_f32(
      /*neg_a=*/false, a, /*neg_b=*/false, o,
      /*c_mod=*/(short)0, c, /*reuse_a=*/false, /*reuse_b=*/false);
  float s = d[0] + d[1] + d[2] + d[3] + d[4] + d[5] + d[6] + d[7];
  #pragma unroll
  for (int m = 16; m >= 1; m >>= 1) s += __shfl_xor(s, m, 32);
  return s * 0.0625f;  // /16, exact
}

// ---------------- Pass A: per-object argmax over priors (chunk partials) ----
__global__ void mbl_passA(const float* __restrict__ boxes,
                          const float* __restrict__ priors,
                          float* __restrict__ pav, int* __restrict__ pai) {
  const int b = blockIdx.y;
  const int tid = threadIdx.x;

  __shared__ float blo[MM][3], bhi[MM][3], bva[MM];
  if (tid < MM) {
    const float* bx = boxes + ((size_t)b * MM + tid) * 6;
    blo[tid][0] = bx[0]; blo[tid][1] = bx[1]; blo[tid][2] = bx[2];
    bhi[tid][0] = bx[3]; bhi[tid][1] = bx[4]; bhi[tid][2] = bx[5];
    bva[tid] = (bx[3] - bx[0]) * (bx[4] - bx[1]) * (bx[5] - bx[2]);
  }
  __syncthreads();

  float bv[MM];
  int   bi[MM];
  #pragma unroll
  for (int m = 0; m < MM; ++m) { bv[m] = -1.0f; bi[m] = 0x7fffffff; }

  const int base = blockIdx.x * CHUNK;
  for (int k = 0; k < PT; ++k) {
    const int p = base + k * BLK + tid;
    // priors row = 24 bytes at p*24 -> always 8-byte aligned: three b64 loads
    const float2* pr2 = reinterpret_cast<const float2*>(priors + (size_t)p * 6);
    const float2 a0 = pr2[0], a1 = pr2[1], a2 = pr2[2];
    const float pc0 = a0.x, pc1 = a0.y, pc2 = a1.x;
    const float ps0 = a1.y, ps1 = a2.x, ps2 = a2.y;
    const float plo0 = pc0 - ps0 * 0.5f, plo1 = pc1 - ps1 * 0.5f, plo2 = pc2 - ps2 * 0.5f;
    const float phi0 = pc0 + ps0 * 0.5f, phi1 = pc1 + ps1 * 0.5f, phi2 = pc2 + ps2 * 0.5f;
    const float vb = (phi0 - plo0) * (phi1 - plo1) * (phi2 - plo2);
    #pragma unroll
    for (int m = 0; m < MM; ++m) {
      const float d0 = fmaxf(fminf(bhi[m][0], phi0) - fmaxf(blo[m][0], plo0), 0.0f);
      const float d1 = fmaxf(fminf(bhi[m][1], phi1) - fmaxf(blo[m][1], plo1), 0.0f);
      const float d2 = fmaxf(fminf(bhi[m][2], phi2) - fmaxf(blo[m][2], plo2), 0.0f);
      const float inter = d0 * d1 * d2;
      const float iou = inter / (bva[m] + vb - inter);
      if (iou > bv[m]) { bv[m] = iou; bi[m] = p; }  // strict >: keep earlier p
    }
  }

  __shared__ float sv[MM * BLK];
  __shared__ int   si[MM * BLK];
  #pragma unroll
  for (int m = 0; m < MM; ++m) { sv[m * BLK + tid] = bv[m]; si[m * BLK + tid] = bi[m]; }
  __syncthreads();
  for (int s = BLK / 2; s > 0; s >>= 1) {
    if (tid < s) {
      #pragma unroll
      for (int m = 0; m < MM; ++m) {
        const float v1 = sv[m * BLK + tid], v2 = sv[m * BLK + tid + s];
        const int   i1 = si[m * BLK + tid], i2 = si[m * BLK + tid + s];
        if (v2 > v1 || (v2 == v1 && i2 < i1)) { sv[m * BLK + tid] = v2; si[m * BLK + tid] = i2; }
      }
    }
    __syncthreads();
  }
  if (tid < MM) {
    const int ob = b * gridDim.x + blockIdx.x;
    pav[(size_t)ob * MM + tid] = sv[tid * BLK];
    pai[(size_t)ob * MM + tid] = si[tid * BLK];
  }
}

// ---------------- Pass A2: combine chunk partials -> prior_feo --------------
__global__ void mbl_feo(const float* __restrict__ pav, const int* __restrict__ pai,
                        int* __restrict__ feo) {
  const int t = threadIdx.x;
  if (t >= BB * MM) return;
  const int b = t / MM, m = t % MM;
  float bv = -1.0f; int bi = 0x7fffffff;
  for (int j = 0; j < NBLK_X; ++j) {               // ascending chunk order
    const int ob = b * NBLK_X + j;
    const float v = pav[(size_t)ob * MM + m];
    const int   i = pai[(size_t)ob * MM + m];
    if (v > bv || (v == bv && i < bi)) { bv = v; bi = i; }
  }
  feo[b * MM + m] = bi;
}

// ---------------- Pass B: matching + focal + L1, WMMA reduction -------------
__global__ void mbl_passB(const float* __restrict__ locs,
                          const float* __restrict__ scores,
                          const float* __restrict__ boxes,
                          const int* __restrict__ labels,
                          const float* __restrict__ priors,
                          const int* __restrict__ feo,
                          float* __restrict__ pb) {
  const int b = blockIdx.y;
  const int tid = threadIdx.x;

  __shared__ float blo[MM][3], bhi[MM][3], bva[MM];
  __shared__ int   slab[MM], sfeo[MM];
  if (tid < MM) {
    const float* bx = boxes + ((size_t)b * MM + tid) * 6;
    blo[tid][0] = bx[0]; blo[tid][1] = bx[1]; blo[tid][2] = bx[2];
    bhi[tid][0] = bx[3]; bhi[tid][1] = bx[4]; bhi[tid][2] = bx[5];
    bva[tid] = (bx[3] - bx[0]) * (bx[4] - bx[1]) * (bx[5] - bx[2]);
    slab[tid] = labels[b * MM + tid];
    sfeo[tid] = feo[b * MM + tid];
  }
  __syncthreads();

  float locsum = 0.0f, focsum = 0.0f, nposf = 0.0f;
  const int base = blockIdx.x * CHUNK;

  for (int k = 0; k < PT; ++k) {
    const int p = base + k * BLK + tid;
    const float2* pr2 = reinterpret_cast<const float2*>(priors + (size_t)p * 6);
    const float2 a0 = pr2[0], a1 = pr2[1], a2 = pr2[2];
    const float pc0 = a0.x, pc1 = a0.y, pc2 = a1.x;
    const float ps0 = a1.y, ps1 = a2.x, ps2 = a2.y;
    const float plo0 = pc0 - ps0 * 0.5f, plo1 = pc1 - ps1 * 0.5f, plo2 = pc2 - ps2 * 0.5f;
    const float phi0 = pc0 + ps0 * 0.5f, phi1 = pc1 + ps1 * 0.5f, phi2 = pc2 + ps2 * 0.5f;
    const float vb = (phi0 - plo0) * (phi1 - plo1) * (phi2 - plo2);

    float bestv = -1.0f; int bestm = 0;
    #pragma unroll
    for (int m = 0; m < MM; ++m) {
      const float d0 = fmaxf(fminf(bhi[m][0], phi0) - fmaxf(blo[m][0], plo0), 0.0f);
      const float d1 = fmaxf(fminf(bhi[m][1], phi1) - fmaxf(blo[m][1], plo1), 0.0f);
      const float d2 = fmaxf(fminf(bhi[m][2], phi2) - fmaxf(blo[m][2], plo2), 0.0f);
      const float inter = d0 * d1 * d2;
      const float iou = inter / (bva[m] + vb - inter);
      if (iou > bestv) { bestv = iou; bestm = m; }   // first max over m
    }
    // Force-assign: each object owns its best prior (later m wins on dup).
    #pragma unroll
    for (int m = 0; m < MM; ++m) {
      if (sfeo[m] == p) { bestm = m; bestv = 1.0f; }
    }
    int lab = slab[bestm];
    if (bestv < THRESH) lab = 0;

    // sigmoid focal loss on foreground logit (8-byte vector load of both logits)
    const float2 sc = reinterpret_cast<const float2*>(scores)[(size_t)b * PP + p];
    const float x = sc.y;
    const float l1p = log1pf(expf(-fabsf(x)));
    const float sp_pos = fmaxf(-x, 0.0f) + l1p;  // softplus(-x) = CE for t=1
    const float sp_neg = fmaxf(x, 0.0f) + l1p;   // softplus(x)  = CE for t=0
    const float prob = 1.0f / (1.0f + expf(-x));
    focsum += (lab > 0) ? ALPHA_W * (1.0f - prob) * (1.0f - prob) * sp_pos
                        : (1.0f - ALPHA_W) * prob * prob * sp_neg;

    if (lab > 0) {
      nposf += 1.0f;
      const float gl0 = blo[bestm][0], gl1 = blo[bestm][1], gl2 = blo[bestm][2];
      const float gh0 = bhi[bestm][0], gh1 = bhi[bestm][1], gh2 = bhi[bestm][2];
      const float gc0 = (gl0 + gh0) * 0.5f, gc1 = (gl1 + gh1) * 0.5f, gc2 = (gl2 + gh2) * 0.5f;
      const float gs0 = gh0 - gl0, gs1 = gh1 - gl1, gs2 = gh2 - gl2;
      const float t0 = (gc0 - pc0) / (ps0 / 10.0f);
      const float t1 = (gc1 - pc1) / (ps1 / 10.0f);
      const float t2 = (gc2 - pc2) / (ps2 / 10.0f);
      const float t3 = logf(gs0 / ps0) * 5.0f;
      const float t4 = logf(gs1 / ps1) * 5.0f;
      const float t5 = logf(gs2 / ps2) * 5.0f;
      const float2* pl2 = reinterpret_cast<const float2*>(locs + ((size_t)b * PP + p) * 6);
      const float2 l0 = pl2[0], l1 = pl2[1], l2 = pl2[2];
      locsum += fabsf(l0.x - t0) + fabsf(l0.y - t1) + fabsf(l1.x - t2) +
                fabsf(l1.y - t3) + fabsf(l2.x - t4) + fabsf(l2.y - t5);
    }
  }

  // Wave-level exact f32 reductions via WMMA (uniform control flow here).
  const float L = wave_allreduce_sum(locsum);
  const float F = wave_allreduce_sum(focsum);
  const float N = wave_allreduce_sum(nposf);

  __shared__ float acc[3][BLK / 32];
  const int w = tid >> 5;
  if ((tid & 31) == 0) { acc[0][w] = L; acc[1][w] = F; acc[2][w] = N; }
  __syncthreads();
  if (tid == 0) {
    float l = 0.0f, f = 0.0f, n = 0.0f;
    #pragma unroll
    for (int j = 0; j < BLK / 32; ++j) { l += acc[0][j]; f += acc[1][j]; n += acc[2][j]; }
    const int ob = b * gridDim.x + blockIdx.x;
    pb[(size_t)ob * 3 + 0] = l;
    pb[(size_t)ob * 3 + 1] = f;
    pb[(size_t)ob * 3 + 2] = n;
  }
}

// ---------------- Finalize: deterministic serial reduction ------------------
__global__ void mbl_finalize(const float* __restrict__ pb, float* __restrict__ out) {
  __shared__ float tot[3];
  const int t = threadIdx.x;
  if (t < 3) {
    float s = 0.0f;
    for (int j = 0; j < NPARTIAL; ++j) s += pb[(size_t)j * 3 + t];
    tot[t] = s;
  }
  __syncthreads();
  if (t == 0) {
    const float np = tot[2];
    out[0] = tot[1] / (float)((size_t)BB * PP);                 // conf_loss (mean focal)
    out[1] = (np > 0.0f) ? tot[0] / fmaxf(np * 6.0f, 1.0f)      // loc_loss
                         : 0.0f;
  }
}

extern "C" void kernel_launch(void* const* d_in, const int* in_sizes, int n_in,
                              void* d_out, int out_size, void* d_ws, size_t ws_size,
                              hipStream_t stream) {
  const float* locs   = (const float*)d_in[0];  // [B,P,6]
  const float* scores = (const float*)d_in[1];  // [B,P,2]
  const float* boxes  = (const float*)d_in[2];  // [B,M,6]
  const int*   labels = (const int*)d_in[3];    // [B,M]
  const float* priors = (const float*)d_in[4];  // [P,6]

  char* ws = (char*)d_ws;
  float* pav = (float*)(ws);                         // NPARTIAL*MM floats (32 KB)
  int*   pai = (int*)(ws + 32768);                   // NPARTIAL*MM ints   (32 KB)
  int*   feo = (int*)(ws + 65536);                   // BB*MM ints         (0.5 KB)
  float* pb  = (float*)(ws + 66560);                 // NPARTIAL*3 floats  (6 KB)
  float* out = (float*)d_out;

  mbl_passA<<<dim3(NBLK_X, BB), BLK, 0, stream>>>(boxes, priors, pav, pai);
  mbl_feo<<<1, 128, 0, stream>>>(pav, pai, feo);
  mbl_passB<<<dim3(NBLK_X, BB), BLK, 0, stream>>>(locs, scores, boxes, labels, priors, feo, pb);
  mbl_finalize<<<1, 32, 0, stream>>>(pb, out);
}